// Confidence_Loss_2_33131377721742
// MI455X (gfx1250) — compile-verified
//
#include <hip/hip_runtime.h>

// ---------------------------------------------------------------------------
// Confidence loss (offset-sampled label agreement) for MI455X / gfx1250.
//   B=16, H=512, W=1024. Bandwidth-bound (~128MB streamed + L2-resident gather).
// Pass 1: vectorized elementwise + gather + per-block fp32 partial sums.
// Pass 2: single wave32 reduces 2048 partials via V_WMMA_F32_16X16X4_F32
//         (B = ones => row sums, full fp32 accumulation in the matrix unit).
// ---------------------------------------------------------------------------

typedef __attribute__((ext_vector_type(2))) float v2f;
typedef __attribute__((ext_vector_type(8))) float v8f;

constexpr int   kH        = 512;
constexpr int   kW        = 1024;
constexpr int   kHW       = kH * kW;
constexpr float kEps      = 1e-7f;
constexpr int   kIgnore   = -1;
constexpr int   kBlocks   = 2048;   // partial count; multiple of 64 for WMMA pass
constexpr int   kThreads  = 256;    // 8 waves of 32 on CDNA5

__global__ __launch_bounds__(kThreads)
void conf_loss_partial(const float* __restrict__ offset,
                       const float* __restrict__ f,
                       const int*   __restrict__ target,
                       float*       __restrict__ partial,
                       int B) {
  const int nvec = (B * kHW) >> 2;               // float4 work items (fits int32)
  float acc = 0.0f;

  for (int v = blockIdx.x * kThreads + threadIdx.x; v < nvec;
       v += gridDim.x * kThreads) {
    const int idx = v << 2;                      // element index, multiple of 4
    const int b   = idx / kHW;
    const int rem = idx - b * kHW;               // position within plane
    const int y   = rem / kW;
    const int x   = rem % kW;                    // multiple of 4 -> no row wrap

    const int   planeOff = b * kHW;
    const int*  tplane   = target + planeOff;

    // 16-byte vector loads (global_load_b128)
    const float4 ox4 = *(const float4*)(offset + (b * 2    ) * kHW + rem);
    const float4 oy4 = *(const float4*)(offset + (b * 2 + 1) * kHW + rem);
    const float4 f4  = *(const float4*)(f + planeOff + rem);
    const int4   t4  = *(const int4*)(tplane + rem);

    float oxa[4]; *(float4*)oxa = ox4;
    float oya[4]; *(float4*)oya = oy4;
    float fa[4];  *(float4*)fa  = f4;
    int   ta[4];  *(int4*)ta    = t4;

    const float gy = -1.0f + 2.0f * (float)y / (float)(kH - 1);

#pragma unroll
    for (int j = 0; j < 4; ++j) {
      const float gx = -1.0f + 2.0f * (float)(x + j) / (float)(kW - 1);
      // grid_sample align_corners=false unnormalization, nearest + border clamp
      const float ix = ((gx + oxa[j] + 1.0f) * (float)kW - 1.0f) * 0.5f;
      const float iy = ((gy + oya[j] + 1.0f) * (float)kH - 1.0f) * 0.5f;
      const int ixn = (int)fminf(fmaxf(rintf(ix), 0.0f), (float)(kW - 1));
      const int iyn = (int)fminf(fmaxf(rintf(iy), 0.0f), (float)(kH - 1));

      int ts = tplane[iyn * kW + ixn];           // L2-resident gather
      ts = (ts == kIgnore) ? 0 : ts;
      int tl = ta[j];
      tl = (tl == kIgnore) ? 0 : tl;

      const float fv  = fa[j];
      const float arg = (tl == ts) ? (fv + kEps) : (1.0f - fv + kEps);
      acc -= __logf(arg);
    }
  }

  // wave32 reduction
#pragma unroll
  for (int off = 16; off > 0; off >>= 1)
    acc += __shfl_down(acc, off, 32);

  __shared__ float lds[kThreads / 32];
  const int lane = threadIdx.x & 31;
  const int wave = threadIdx.x >> 5;
  if (lane == 0) lds[wave] = acc;
  __syncthreads();

  if (wave == 0) {
    float s = (lane < kThreads / 32) ? lds[lane] : 0.0f;
#pragma unroll
    for (int off = 4; off > 0; off >>= 1)
      s += __shfl_down(s, off, 32);
    if (lane == 0) partial[blockIdx.x] = s;
  }
}

// Single-wave final reduction: 2048 partials summed through the matrix unit.
// A (16x4 f32) holds 64 partials/iteration, B = all-ones => D accumulates row
// sums (fp32). Total = s(lane0) + s(lane16) per the 16x16 f32 C/D layout.
__global__ __launch_bounds__(32)
void conf_loss_final(const float* __restrict__ partial,
                     float* __restrict__ out,
                     float scale) {
  const int lane = threadIdx.x;                  // 32 lanes, EXEC all-1s
  v8f c = {0.f, 0.f, 0.f, 0.f, 0.f, 0.f, 0.f, 0.f};
  const v2f ones = {1.0f, 1.0f};

#pragma unroll 4
  for (int i = 0; i < kBlocks / 64; ++i) {
    v2f a;
    a.x = partial[i * 64 + lane * 2 + 0];
    a.y = partial[i * 64 + lane * 2 + 1];
    // v_wmma_f32_16x16x4_f32: D = A x ones + C  (row sums, fp32 accumulate)
    c = __builtin_amdgcn_wmma_f32_16x16x4_f32(
        /*neg_a=*/false, a, /*neg_b=*/false, ones,
        /*c_mod=*/(short)0, c, /*reuse_a=*/false, /*reuse_b=*/false);
  }

  float s = c[0] + c[1] + c[2] + c[3] + c[4] + c[5] + c[6] + c[7];
  const float total = __shfl(s, 0, 32) + __shfl(s, 16, 32);
  if (lane == 0) out[0] = total * scale;
}

extern "C" void kernel_launch(void* const* d_in, const int* in_sizes, int n_in,
                              void* d_out, int out_size, void* d_ws, size_t ws_size,
                              hipStream_t stream) {
  const float* offset = (const float*)d_in[0];   // (B,2,H,W) f32
  const float* f      = (const float*)d_in[1];   // (B,1,H,W) f32
  const int*   target = (const int*)d_in[2];     // (B,H,W)   i32

  const int B = in_sizes[2] / kHW;               // 16 for the reference shapes
  float* partial = (float*)d_ws;                 // kBlocks floats = 8 KB

  conf_loss_partial<<<kBlocks, kThreads, 0, stream>>>(offset, f, target,
                                                      partial, B);
  conf_loss_final<<<1, 32, 0, stream>>>(partial, (float*)d_out,
                                        1.0f / (float)kHW);
}